// FeatureInterpolator_1717986918815
// MI455X (gfx1250) — compile-verified
//
#include <hip/hip_runtime.h>

typedef __attribute__((ext_vector_type(2))) float v2f;
typedef __attribute__((ext_vector_type(8))) float v8f;

// Problem shape (fixed by the reference)
constexpr int B   = 8;
constexpr int N1  = 8192;
constexpr int N2  = 1024;
constexpr int C1  = 128;
constexpr int C2  = 256;
constexpr int CO  = C1 + C2;   // 384 output channels
constexpr int TM  = 16;        // n1 points per workgroup (one WMMA M-tile)
constexpr int CHUNK = 512;     // n2 columns resident in LDS at a time

__global__ __launch_bounds__(256) void fi3nn_kernel(
    const float* __restrict__ xyz1,   // (B,3,N1)
    const float* __restrict__ xyz2,   // (B,3,N2)
    const float* __restrict__ f1,     // (B,C1,N1)
    const float* __restrict__ f2,     // (B,C2,N2)
    float* __restrict__ out)          // (B,CO,N1)
{
    __shared__ float distT[CHUNK * TM];          // 32 KB, [n_local][m]
    __shared__ float cand_d[TM * 16 * 3];
    __shared__ int   cand_i[TM * 16 * 3];
    __shared__ float wRow[TM][3];
    __shared__ int   iRow[TM][3];

    const int tid    = threadIdx.x;
    const int bid    = blockIdx.x;
    const int b      = bid / (N1 / TM);
    const int n1base = (bid % (N1 / TM)) * TM;

    const int lane = tid & 31;
    const int wave = tid >> 5;          // 0..7
    const int hi16 = lane >> 4;         // lane half: selects K=0,1 vs K=2,3
    const int l16  = lane & 15;

    const float* p1 = xyz1 + b * 3 * N1;
    const float* p2 = xyz2 + b * 3 * N2;

    // ---- A matrix (16x4 f32): row m = (-2x1, -2y1, -2z1, 1) ----
    const int   am = n1base + l16;
    const float x1 = p1[0 * N1 + am];
    const float y1 = p1[1 * N1 + am];
    const float z1 = p1[2 * N1 + am];
    v2f a;
    a.x = hi16 ? (-2.0f * z1) : (-2.0f * x1);
    a.y = hi16 ? 1.0f         : (-2.0f * y1);

    // Running top-3 (smallest partial dist) per thread; persists across chunks.
    const int srow = tid >> 4;   // row m this thread scans   (0..15)
    const int scol = tid & 15;   // scan lane within the row  (0..15)
    float d0 = 3.0e38f, d1 = 3.0e38f, d2 = 3.0e38f;
    int   i0 = N2,      i1 = N2,      i2 = N2;

    for (int ch = 0; ch < N2 / CHUNK; ++ch) {
        // ---- Phase 1: WMMA distance tiles into LDS ----
        for (int t = 0; t < CHUNK / (8 * 16); ++t) {       // 4 tiles per wave
            const int nloc = (wave * 4 + t) * 16 + l16;    // 0..511
            const int n    = ch * CHUNK + nloc;
            const float x2 = p2[0 * N2 + n];
            const float y2 = p2[1 * N2 + n];
            const float z2 = p2[2 * N2 + n];
            v2f bb;
            bb.x = hi16 ? z2 : x2;
            bb.y = hi16 ? (x2 * x2 + y2 * y2 + z2 * z2) : y2;
            v8f c = {};
            // D[m][n] = -2*(p1 . p2) + |p2|^2   (|p1|^2 added later; row-const)
            v8f d = __builtin_amdgcn_wmma_f32_16x16x4_f32(
                false, a, false, bb, (short)0, c, false, false);
            // D layout: vgpr r holds M = r + 8*hi16, N = nloc. Store transposed
            // [n][m] so the 8 values are contiguous -> two ds_store_b128.
            float* dst = &distT[nloc * TM + hi16 * 8];
            *(float4*)(dst + 0) = make_float4(d[0], d[1], d[2], d[3]);
            *(float4*)(dst + 4) = make_float4(d[4], d[5], d[6], d[7]);
        }
        __syncthreads();

        // ---- Phase 2a: strided running min-3 per (row, scan-lane) ----
        for (int k = 0; k < CHUNK / 16; ++k) {
            const int nloc = scol + k * 16;
            const float v  = distT[nloc * TM + srow];
            const int   n  = ch * CHUNK + nloc;
            if (v < d0)      { d2 = d1; i2 = i1; d1 = d0; i1 = i0; d0 = v; i0 = n; }
            else if (v < d1) { d2 = d1; i2 = i1; d1 = v;  i1 = n; }
            else if (v < d2) { d2 = v;  i2 = n; }
        }
        __syncthreads();   // distT reusable for next chunk
    }

    {
        const int cb = (srow * 16 + scol) * 3;
        cand_d[cb + 0] = d0; cand_d[cb + 1] = d1; cand_d[cb + 2] = d2;
        cand_i[cb + 0] = i0; cand_i[cb + 1] = i1; cand_i[cb + 2] = i2;
    }
    __syncthreads();

    // ---- Phase 2b: one thread per row merges 48 candidates, builds weights ----
    if (tid < TM) {
        float e0 = 3.0e38f, e1 = 3.0e38f, e2 = 3.0e38f;
        int   j0 = N2,      j1 = N2,      j2 = N2;
        for (int s = 0; s < 16; ++s) {
            for (int q = 0; q < 3; ++q) {
                const int   cb = (tid * 16 + s) * 3 + q;
                const float v  = cand_d[cb];
                const int   i  = cand_i[cb];
                const bool b0 = (v < e0) || ((v == e0) && (i < j0));
                const bool b1 = (v < e1) || ((v == e1) && (i < j1));
                const bool b2 = (v < e2) || ((v == e2) && (i < j2));
                if (b0)      { e2 = e1; j2 = j1; e1 = e0; j1 = j0; e0 = v; j0 = i; }
                else if (b1) { e2 = e1; j2 = j1; e1 = v;  j1 = i; }
                else if (b2) { e2 = v;  j2 = i; }
            }
        }
        const int   mm  = n1base + tid;
        const float sx  = p1[0 * N1 + mm];
        const float sy  = p1[1 * N1 + mm];
        const float sz  = p1[2 * N1 + mm];
        const float sq1 = sx * sx + sy * sy + sz * sz;
        const float dd0 = fmaxf(e0 + sq1, 1e-10f);
        const float dd1 = fmaxf(e1 + sq1, 1e-10f);
        const float dd2 = fmaxf(e2 + sq1, 1e-10f);
        const float r0 = 1.0f / dd0, r1 = 1.0f / dd1, r2 = 1.0f / dd2;
        const float inv_sum = 1.0f / (r0 + r1 + r2);
        wRow[tid][0] = r0 * inv_sum; iRow[tid][0] = j0;
        wRow[tid][1] = r1 * inv_sum; iRow[tid][1] = j1;
        wRow[tid][2] = r2 * inv_sum; iRow[tid][2] = j2;
    }
    __syncthreads();

    // ---- Phase 3: interpolate C2 channels (coalesced writes over n1) ----
    {
        const int m  = tid & 15;        // n1 offset -> contiguous across lanes
        const int c0 = tid >> 4;        // 0..15
        const float w0 = wRow[m][0], w1 = wRow[m][1], w2 = wRow[m][2];
        const int   g0 = iRow[m][0], g1 = iRow[m][1], g2 = iRow[m][2];
        const float* f2b  = f2 + b * C2 * N2;
        float*       outb = out + (size_t)b * CO * N1 + n1base + m;
        for (int cc = 0; cc < C2 / 16; ++cc) {
            const int c = cc * 16 + c0;
            const float* row = f2b + c * N2;
            outb[(size_t)c * N1] = w0 * row[g0] + w1 * row[g1] + w2 * row[g2];
        }
    }

    // ---- Phase 4: concat features1 into channels [C2, C2+C1) ----
    {
        const int m  = tid & 15;
        const int c0 = tid >> 4;
        const float* f1b  = f1 + (size_t)b * C1 * N1 + n1base + m;
        float*       outb = out + (size_t)b * CO * N1 + (size_t)C2 * N1 + n1base + m;
        for (int cc = 0; cc < C1 / 16; ++cc) {
            const int c = cc * 16 + c0;
            outb[(size_t)c * N1] = f1b[(size_t)c * N1];
        }
    }
}

extern "C" void kernel_launch(void* const* d_in, const int* in_sizes, int n_in,
                              void* d_out, int out_size, void* d_ws, size_t ws_size,
                              hipStream_t stream) {
    const float* xyz1 = (const float*)d_in[0];
    const float* xyz2 = (const float*)d_in[1];
    const float* f1   = (const float*)d_in[2];
    const float* f2   = (const float*)d_in[3];
    float* out        = (float*)d_out;

    const int blocks = B * (N1 / TM);   // 4096
    fi3nn_kernel<<<blocks, 256, 0, stream>>>(xyz1, xyz2, f1, f2, out);
}